// Net_88210038326466
// MI455X (gfx1250) — compile-verified
//
#include <hip/hip_runtime.h>
#include <hip/hip_bf16.h>

// ---------------------------------------------------------------------------
// AGNN (2-layer) forward for MI455X / gfx1250, wave32.
//   h1 = relu(x @ W1 + b1)                  -> WMMA f32 16x16x4 GEMM (TDM-staged W1)
//   h2 = agnn_prop(h1, beta=1)              -> edge kernels, fp32 atomics
//   h3 = agnn_prop(h2, beta=beta2)
//   out = log_softmax(h3 @ W2 + b2)
// segment_max is replaced by the exact bound |beta| (cosine in [-1,1]); the
// shift cancels in ex/denom, so results are identical up to fp rounding.
// ---------------------------------------------------------------------------

#define F_IN 512
#define HID  16
#define NCLS 40

typedef __attribute__((ext_vector_type(2))) float        v2f;
typedef __attribute__((ext_vector_type(8))) float        v8f;
typedef __attribute__((ext_vector_type(4))) unsigned int v4u;
typedef __attribute__((ext_vector_type(4))) int          v4i;
typedef __attribute__((ext_vector_type(8))) int          v8i;

// ---------------------------------------------------------------------------
// Kernel 1: h = relu(x @ W1 + b1) using V_WMMA_F32_16X16X4_F32.
// One wave handles a 16-row tile (HID=16 == one N tile). 8 waves / block.
// W1 (512x16 f32 = 32KB) is staged into LDS by a single TDM descriptor
// (tensor_load_to_lds, TENSORcnt), issued by wave 0.
//
// A (16x4 f32) lane layout:  lanes 0-15: VGPR0=K0, VGPR1=K1 ; lanes 16-31: K2,K3
// B (4x16 f32) mirror:       VGPR0 = row K0/K2, VGPR1 = row K1/K3
// C/D (16x16 f32): VGPR r, lanes 0-15 -> (M=r, N=lane); lanes 16-31 -> (M=r+8)
// ---------------------------------------------------------------------------
__global__ __launch_bounds__(256) void gemm1_relu_wmma(
    const float* __restrict__ x, const float* __restrict__ W1,
    const float* __restrict__ b1, float* __restrict__ h, int n_nodes)
{
    __shared__ float lds_w[F_IN * HID];  // 32 KB

#if defined(__has_builtin) && __has_builtin(__builtin_amdgcn_tensor_load_to_lds)
    if ((threadIdx.x >> 5) == 0) {
        // ---- Tensor Data Mover: copy W1 (8192 f32) -> LDS as one 2D tile ----
        const unsigned long long ga = (unsigned long long)(size_t)W1;
        const unsigned int lbase    = (unsigned int)(size_t)(void*)lds_w;
        const unsigned int dim0     = F_IN * HID;              // 8192 elements
        v4u g0;
        g0.x = 0x1u;                                           // count=1 (valid), user D#
        g0.y = lbase;                                          // lds_addr[31:0]
        g0.z = (unsigned int)ga;                               // global_addr[31:0]
        g0.w = (unsigned int)((ga >> 32) & 0x01FFFFFFu)        // global_addr[56:32]
             | 0x80000000u;                                    // type = 2 ("image")
        v8i g1;
        g1[0] = (int)(2u << 16);                               // data_size = 4 bytes
        g1[1] = (int)((dim0 & 0xFFFFu) << 16);                 // tensor_dim0[15:0]
        g1[2] = (int)((dim0 >> 16) | (1u << 16));              // tensor_dim0[31:16] | tensor_dim1=1
        g1[3] = (int)((dim0 & 0xFFFFu) << 16);                 // tile_dim0 = 8192
        g1[4] = 1;                                             // tile_dim1 = 1, tile_dim2 = 0
        g1[5] = (int)dim0;                                     // tensor_dim0_stride[31:0]
        g1[6] = 0;                                             // stride hi / dim1_stride lo
        g1[7] = 0;
        v4i z4 = {0, 0, 0, 0};
#if __clang_major__ >= 23
        v8i z8 = {0, 0, 0, 0, 0, 0, 0, 0};
        __builtin_amdgcn_tensor_load_to_lds(g0, g1, z4, z4, z8, 0);
#else
        __builtin_amdgcn_tensor_load_to_lds(g0, g1, z4, z4, 0);
#endif
        __builtin_amdgcn_s_wait_tensorcnt(0);                  // s_wait_tensorcnt 0
    }
#else
    {   // fallback: cooperative vector copy
        const float4* w4 = (const float4*)W1;
        float4*       l4 = (float4*)lds_w;
        for (int i = threadIdx.x; i < (F_IN * HID) / 4; i += 256)
            l4[i] = w4[i];
    }
#endif
    __syncthreads();

    const int wave = threadIdx.x >> 5;
    const int lane = threadIdx.x & 31;
    const int half = lane >> 4;      // 0: lanes 0-15, 1: lanes 16-31
    const int m    = lane & 15;

    const long tile = (long)blockIdx.x * 8 + wave;
    const long row0 = tile * 16;
    if (row0 >= n_nodes) return;     // wave-uniform -> EXEC stays all-ones

    const long row  = row0 + m;
    const long rowc = (row < n_nodes) ? row : (n_nodes - 1);  // clamp (EXEC full)
    const float* xrow = x + rowc * F_IN;

    const float bias = b1[m];
    v8f c = {bias, bias, bias, bias, bias, bias, bias, bias};

    #pragma unroll 4
    for (int k0 = 0; k0 < F_IN; k0 += 4) {
        const int kk = k0 + 2 * half;
        const float2 av = *(const float2*)(xrow + kk);        // A frag (b64 load)
        v2f a; a.x = av.x; a.y = av.y;
        v2f b; b.x = lds_w[kk * HID + m];                     // B frag (ds loads)
               b.y = lds_w[(kk + 1) * HID + m];
        c = __builtin_amdgcn_wmma_f32_16x16x4_f32(
                /*neg_a=*/false, a, /*neg_b=*/false, b,
                /*c_mod=*/(short)0, c, /*reuse_a=*/false, /*reuse_b=*/false);
    }

    // write back with relu: VGPR r -> row0 + 8*half + r, col m
    float* hp = h + (row0 + 8 * half) * HID + m;
    if (row0 + 16 <= n_nodes) {      // full tile: unconditional coalesced stores
        #pragma unroll
        for (int r = 0; r < 8; ++r) {
            float v = c[r];
            hp[r * HID] = v > 0.f ? v : 0.f;
        }
    } else {                          // tail tile: guarded
        #pragma unroll
        for (int r = 0; r < 8; ++r) {
            if (row0 + 8 * half + r < n_nodes) {
                float v = c[r];
                hp[r * HID] = v > 0.f ? v : 0.f;
            }
        }
    }
}

// ---------------------------------------------------------------------------
// Kernel 2: row-wise L2 normalize (16 features / node), one thread per node.
// ---------------------------------------------------------------------------
__global__ __launch_bounds__(256) void l2norm16(
    const float* __restrict__ h, float* __restrict__ xn, int n_nodes)
{
    const int i = blockIdx.x * blockDim.x + threadIdx.x;
    if (i >= n_nodes) return;
    const float4* hp = (const float4*)(h + (long)i * HID);
    float4 a = hp[0], b = hp[1], c = hp[2], d = hp[3];
    float ss = a.x*a.x + a.y*a.y + a.z*a.z + a.w*a.w
             + b.x*b.x + b.y*b.y + b.z*b.z + b.w*b.w
             + c.x*c.x + c.y*c.y + c.z*c.z + c.w*c.w
             + d.x*d.x + d.y*d.y + d.z*d.z + d.w*d.w;
    const float inv = 1.0f / fmaxf(sqrtf(ss), 1e-12f);
    float4* op = (float4*)(xn + (long)i * HID);
    a.x*=inv; a.y*=inv; a.z*=inv; a.w*=inv;
    b.x*=inv; b.y*=inv; b.z*=inv; b.w*=inv;
    c.x*=inv; c.y*=inv; c.z*=inv; c.w*=inv;
    d.x*=inv; d.y*=inv; d.z*=inv; d.w*=inv;
    op[0]=a; op[1]=b; op[2]=c; op[3]=d;
}

// ---------------------------------------------------------------------------
// Kernel 3: per-edge ex = exp(beta*cos(src,dst) - |beta|); denom[dst] += ex.
// 16 lanes per edge -> each node gather is one contiguous 64B line.
// Self-loops are the last n_nodes "virtual" edges (src=dst=i).
// ---------------------------------------------------------------------------
__global__ __launch_bounds__(256) void edge_softmax_num(
    const float* __restrict__ xn, const int* __restrict__ ei,
    float* __restrict__ exbuf, float* __restrict__ denom,
    const float* __restrict__ beta_ptr, long n_edges_raw, long n_edges_tot)
{
    const int lane16 = threadIdx.x & 15;
    const long e = ((long)blockIdx.x * blockDim.x + threadIdx.x) >> 4;
    if (e >= n_edges_tot) return;   // uniform within a 16-lane group

    int src, dst;
    if (e < n_edges_raw) { src = ei[e]; dst = ei[n_edges_raw + e]; }
    else                 { src = dst = (int)(e - n_edges_raw); }

    const float beta = beta_ptr ? *beta_ptr : 1.0f;
    float p = xn[(long)src * HID + lane16] * xn[(long)dst * HID + lane16];
    p += __shfl_xor(p, 8, 16);
    p += __shfl_xor(p, 4, 16);
    p += __shfl_xor(p, 2, 16);
    p += __shfl_xor(p, 1, 16);

    if (lane16 == 0) {
        const float v = __expf(beta * p - fabsf(beta));  // exact-shift softmax
        exbuf[e] = v;
        atomicAdd(&denom[dst], v);
    }
}

// ---------------------------------------------------------------------------
// Kernel 4: out[dst] += (ex[e]/denom[dst]) * h[src], 16 lanes per edge.
// ---------------------------------------------------------------------------
__global__ __launch_bounds__(256) void edge_aggregate(
    const float* __restrict__ hin, const int* __restrict__ ei,
    const float* __restrict__ exbuf, const float* __restrict__ denom,
    float* __restrict__ out, long n_edges_raw, long n_edges_tot)
{
    const int lane16 = threadIdx.x & 15;
    const long e = ((long)blockIdx.x * blockDim.x + threadIdx.x) >> 4;
    if (e >= n_edges_tot) return;

    int src, dst;
    if (e < n_edges_raw) { src = ei[e]; dst = ei[n_edges_raw + e]; }
    else                 { src = dst = (int)(e - n_edges_raw); }

    float w = exbuf[e] / denom[dst];
    w = __shfl(w, (int)(threadIdx.x & ~15), 32);  // broadcast lane-group leader
    atomicAdd(&out[(long)dst * HID + lane16], w * hin[(long)src * HID + lane16]);
}

// ---------------------------------------------------------------------------
// Kernel 5: logits = h @ W2 + b2 (16x40), then log_softmax; 1 thread / node.
// ---------------------------------------------------------------------------
__global__ __launch_bounds__(256) void head_logsoftmax(
    const float* __restrict__ h, const float* __restrict__ W2,
    const float* __restrict__ b2, float* __restrict__ out, int n_nodes)
{
    __shared__ float lw[HID * NCLS];
    __shared__ float lb[NCLS];
    for (int i = threadIdx.x; i < HID * NCLS; i += 256) lw[i] = W2[i];
    for (int i = threadIdx.x; i < NCLS;       i += 256) lb[i] = b2[i];
    __syncthreads();

    const int node = blockIdx.x * blockDim.x + threadIdx.x;
    if (node >= n_nodes) return;

    float hv[HID];
    #pragma unroll
    for (int f = 0; f < HID; ++f) hv[f] = h[(long)node * HID + f];

    float logits[NCLS];
    float mx = -3.0e38f;
    #pragma unroll
    for (int c = 0; c < NCLS; ++c) {
        float s = lb[c];
        #pragma unroll
        for (int f = 0; f < HID; ++f) s += hv[f] * lw[f * NCLS + c];
        logits[c] = s;
        mx = fmaxf(mx, s);
    }
    float sum = 0.f;
    #pragma unroll
    for (int c = 0; c < NCLS; ++c) sum += __expf(logits[c] - mx);
    const float lse = mx + __logf(sum);
    #pragma unroll
    for (int c = 0; c < NCLS; ++c)
        out[(long)node * NCLS + c] = logits[c] - lse;
}

// ---------------------------------------------------------------------------
extern "C" void kernel_launch(void* const* d_in, const int* in_sizes, int n_in,
                              void* d_out, int out_size, void* d_ws, size_t ws_size,
                              hipStream_t stream)
{
    const float* x     = (const float*)d_in[0];
    const int*   ei    = (const int*)  d_in[1];
    const float* W1    = (const float*)d_in[2];
    const float* b1    = (const float*)d_in[3];
    const float* W2    = (const float*)d_in[4];
    const float* b2    = (const float*)d_in[5];
    const float* beta2 = (const float*)d_in[6];
    float* out = (float*)d_out;

    const int  n_nodes = in_sizes[0] / F_IN;          // 100000
    const long e_raw   = in_sizes[1] / 2;             // 3,200,000
    const long e_tot   = e_raw + n_nodes;             // + self loops

    // workspace carve-up (256B aligned)
    char* ws = (char*)d_ws;
    size_t off = 0;
    auto carve = [&](size_t bytes) -> float* {
        float* p = (float*)(ws + off);
        off += (bytes + 255) & ~(size_t)255;
        return p;
    };
    float* h1    = carve((size_t)n_nodes * HID * 4);
    float* h2    = carve((size_t)n_nodes * HID * 4);
    float* h3    = carve((size_t)n_nodes * HID * 4);
    float* xn    = carve((size_t)n_nodes * HID * 4);
    float* denom = carve((size_t)n_nodes * 4);
    float* exbuf = carve((size_t)e_tot * 4);
    (void)ws_size; (void)n_in; (void)out_size;

    const int tiles      = (n_nodes + 15) / 16;
    const int gemm_grid  = (tiles + 7) / 8;
    const int node_grid  = (n_nodes + 255) / 256;
    const long edge_thr  = e_tot * 16;
    const int  edge_grid = (int)((edge_thr + 255) / 256);

    // ---- layer 1 GEMM + relu (WMMA, TDM-staged weights) ----
    gemm1_relu_wmma<<<gemm_grid, 256, 0, stream>>>(x, W1, b1, h1, n_nodes);

    // ---- prop 1 (beta = 1.0) ----
    l2norm16<<<node_grid, 256, 0, stream>>>(h1, xn, n_nodes);
    hipMemsetAsync(denom, 0, (size_t)n_nodes * 4, stream);
    hipMemsetAsync(h2,    0, (size_t)n_nodes * HID * 4, stream);
    edge_softmax_num<<<edge_grid, 256, 0, stream>>>(xn, ei, exbuf, denom,
                                                    nullptr, e_raw, e_tot);
    edge_aggregate<<<edge_grid, 256, 0, stream>>>(h1, ei, exbuf, denom, h2,
                                                  e_raw, e_tot);

    // ---- prop 2 (beta = beta2, device scalar) ----
    l2norm16<<<node_grid, 256, 0, stream>>>(h2, xn, n_nodes);
    hipMemsetAsync(denom, 0, (size_t)n_nodes * 4, stream);
    hipMemsetAsync(h3,    0, (size_t)n_nodes * HID * 4, stream);
    edge_softmax_num<<<edge_grid, 256, 0, stream>>>(xn, ei, exbuf, denom,
                                                    beta2, e_raw, e_tot);
    edge_aggregate<<<edge_grid, 256, 0, stream>>>(h2, ei, exbuf, denom, h3,
                                                  e_raw, e_tot);

    // ---- head GEMM + log_softmax ----
    head_logsoftmax<<<node_grid, 256, 0, stream>>>(h3, W2, b2, out, n_nodes);
}